// HeadBase_84988812853820
// MI455X (gfx1250) — compile-verified
//
#include <hip/hip_runtime.h>
#include <math.h>

typedef __attribute__((ext_vector_type(2))) float v2f;
typedef __attribute__((ext_vector_type(8))) float v8f;

#define BATCH    128
#define NN       4096
#define MM       128
#define TPB      256
#define NWAVES   (TPB / 32)
#define NTILES   (NN / 16)
#define PERTHR   (NN / TPB)
#define EPS_COS  1e-8f
#define EPS_SHRP 1e-16f

// ---- block-wide reductions over TPB threads (red[] is TPB floats) ----
__device__ __forceinline__ float blk_sum(float v, float* red) {
    const int t = threadIdx.x;
    red[t] = v;
    __syncthreads();
    #pragma unroll
    for (int s = TPB / 2; s > 0; s >>= 1) {
        if (t < s) red[t] += red[t + s];
        __syncthreads();
    }
    float r = red[0];
    __syncthreads();
    return r;
}

__device__ __forceinline__ float blk_max(float v, float* red) {
    const int t = threadIdx.x;
    red[t] = v;
    __syncthreads();
    #pragma unroll
    for (int s = TPB / 2; s > 0; s >>= 1) {
        if (t < s) red[t] = fmaxf(red[t], red[t + s]);
        __syncthreads();
    }
    float r = red[0];
    __syncthreads();
    return r;
}

// One workgroup per batch element. Fused: cosine-sim (WMMA matvec + VALU norms)
// -> softmax -> gate -> circular shift -> sharpen -> normalize.
__global__ __launch_bounds__(TPB) void ntm_head_kernel(
    const float* __restrict__ kvec_g,   // [B,M]
    const float* __restrict__ beta_g,   // [B]
    const float* __restrict__ prevw_g,  // [B,N]
    const float* __restrict__ gate_g,   // [B]
    const float* __restrict__ s_g,      // [B,3]
    const float* __restrict__ gamma_g,  // [B]
    const float* __restrict__ mem_g,    // [B,N,M]
    float* __restrict__ out_g)          // [B,N]
{
    __shared__ float kvec[MM];
    __shared__ float arr[NN];    // staging: beta*sim -> exp -> w_g
    __shared__ float red[TPB];
    __shared__ float sc[6];      // beta, g, gamma, s0, s1, s2

    const int b    = blockIdx.x;
    const int tid  = threadIdx.x;
    const int lane = tid & 31;
    const int wave = tid >> 5;
    const int lrow = lane & 15;  // row within 16-tile owned by this lane
    const int half = lane >> 4;  // K-half (0: K=0,1 ; 1: K=2,3)

    if (tid < MM) kvec[tid] = kvec_g[(size_t)b * MM + tid];
    if (tid == 0) {
        sc[0] = beta_g[b];
        sc[1] = gate_g[b];
        sc[2] = gamma_g[b];
        sc[3] = s_g[b * 3 + 0];
        sc[4] = s_g[b * 3 + 1];
        sc[5] = s_g[b * 3 + 2];
    }
    __syncthreads();

    // ---- ||k|| ----
    float kp     = (tid < MM) ? kvec[tid] * kvec[tid] : 0.0f;
    float kn     = fmaxf(sqrtf(blk_sum(kp, red)), EPS_COS);
    const float sbeta = sc[0], sg = sc[1], sgam = sc[2];
    const float s0 = sc[3], s1 = sc[4], s2 = sc[5];

    const float* memb = mem_g + (size_t)b * NN * MM;

    // ---- Phase 1: dots via V_WMMA_F32_16X16X4_F32, row norms via VALU ----
    // A = k replicated over all 16 rows  (A[i][kk] = kvec[m0+kk]):
    //   lane<16: vgpr0=K0, vgpr1=K1 ; lane>=16: vgpr0=K2, vgpr1=K3
    // B = memory tile (B[kk][j] = mem[n0+j][m0+kk]):
    //   lane = (kk/2)*16 + j, vgpr = kk%2  -> lane loads float2 at (n0+j, m0+2*half)
    // => every D column j equals dots[n0+j]; lane directly owns row n0+(lane%16).
    float lmax = -3.0e38f;
    for (int tile = wave; tile < NTILES; tile += NWAVES) {
        const int row = tile * 16 + lrow;
        const float* rp = memb + (size_t)row * MM + 2 * half;

        if (tile + NWAVES < NTILES)  // stream-ahead hint -> global_prefetch_b8
            __builtin_prefetch(memb + (size_t)(row + NWAVES * 16) * MM + 2 * half, 0, 0);

        v8f c = {};
        float ssq = 0.0f;
        #pragma unroll 8
        for (int mc = 0; mc < MM / 4; ++mc) {
            v2f bfrag = *(const v2f*)(rp + mc * 4);
            v2f afrag;
            afrag.x = kvec[mc * 4 + 2 * half];
            afrag.y = kvec[mc * 4 + 2 * half + 1];
            ssq = __builtin_fmaf(bfrag.x, bfrag.x, ssq);
            ssq = __builtin_fmaf(bfrag.y, bfrag.y, ssq);
            c = __builtin_amdgcn_wmma_f32_16x16x4_f32(
                    /*neg_a=*/false, afrag, /*neg_b=*/false, bfrag,
                    /*c_mod=*/(short)0, c, /*reuse_a=*/false, /*reuse_b=*/false);
        }
        // fold the two K-halves of the sum of squares (lanes l and l^16 share a row)
        float ssq_full = ssq + __shfl_xor(ssq, 16, 32);
        float mn  = fmaxf(sqrtf(ssq_full), EPS_COS);
        float dot = c[0];                       // identical across D rows
        float x   = sbeta * (dot / (kn * mn));  // beta * cosine-sim
        if (half == 0) arr[row] = x;
        lmax = fmaxf(lmax, x);
    }

    // ---- Phase 2: softmax over N ----
    const float xmax = blk_max(lmax, red);
    float lsum = 0.0f;
    #pragma unroll
    for (int i = 0; i < PERTHR; ++i) {
        const int n = tid + i * TPB;
        const float e = expf(arr[n] - xmax);
        arr[n] = e;
        lsum += e;
    }
    const float inv_denom = 1.0f / blk_sum(lsum, red);

    // ---- Phase 3: gated interpolation (in-place, element-owned) ----
    #pragma unroll
    for (int i = 0; i < PERTHR; ++i) {
        const int n = tid + i * TPB;
        const float wc = arr[n] * inv_denom;
        arr[n] = sg * wc + (1.0f - sg) * prevw_g[(size_t)b * NN + n];
    }
    __syncthreads();  // neighbors needed below

    // ---- Phase 4: circular shift (K=3) + sharpen ----
    float wp[PERTHR];
    float lsum2 = 0.0f;
    #pragma unroll
    for (int i = 0; i < PERTHR; ++i) {
        const int n  = tid + i * TPB;
        const float ws = s0 * arr[(n + NN - 1) & (NN - 1)]
                       + s1 * arr[n]
                       + s2 * arr[(n + 1) & (NN - 1)];
        const float p = powf(ws, sgam);
        wp[i] = p;
        lsum2 += p;
    }
    const float invS = 1.0f / (blk_sum(lsum2, red) + EPS_SHRP);
    #pragma unroll
    for (int i = 0; i < PERTHR; ++i)
        out_g[(size_t)b * NN + tid + i * TPB] = wp[i] * invS;
}

extern "C" void kernel_launch(void* const* d_in, const int* in_sizes, int n_in,
                              void* d_out, int out_size, void* d_ws, size_t ws_size,
                              hipStream_t stream) {
    const float* k      = (const float*)d_in[0];
    const float* beta   = (const float*)d_in[1];
    const float* prev_w = (const float*)d_in[2];
    const float* g      = (const float*)d_in[3];
    const float* s      = (const float*)d_in[4];
    const float* gamma  = (const float*)d_in[5];
    const float* memory = (const float*)d_in[6];
    float* out = (float*)d_out;

    const int B = in_sizes[1];  // beta is [B,1]
    ntm_head_kernel<<<B, TPB, 0, stream>>>(k, beta, prev_w, g, s, gamma, memory, out);
}